// PixelShuffleUpsample_15522011808101
// MI455X (gfx1250) — compile-verified
//
#include <hip/hip_runtime.h>

#define B_    4
#define C_    256
#define H_    128
#define W_    128
#define HW_   (H_*W_)
#define NOUT_ 32
#define LWS   33   // padded LDS row stride (floats) to avoid bank conflicts

typedef float v2f __attribute__((ext_vector_type(2)));
typedef float v8f __attribute__((ext_vector_type(8)));

// ---------------------------------------------------------------------------
// Kernel 1: off[m*32 + o] = sum_c x[b,c,h,w] * wgt[o,c] + bias[o]
//   m = b*H*W + h*W + w  (16 consecutive m stay inside one channel plane)
// GEMM M=65536, N=32, K=256 via V_WMMA_F32_16X16X4_F32.
// A (16x4 f32): lanes 0-15 K={0,1} in v[0],v[1]; lanes 16-31 K={2,3}.
// B (4x16 f32): lane n holds column N; half-lanes split K like A.
// D (16x16):    VGPR r -> M=r (lanes 0-15), M=r+8 (lanes 16-31), N=lane&15.
// ---------------------------------------------------------------------------
__global__ void __launch_bounds__(128) offset_gemm_wmma(
    const float* __restrict__ x, const float* __restrict__ wgt,
    const float* __restrict__ bias, float* __restrict__ off) {
  __shared__ float lw[256 * LWS];           // lw[k*LWS + n] = wgt[n*256 + k]
  int tid = threadIdx.x;
  for (int idx = tid; idx < 256 * NOUT_; idx += 128) {
    int k = idx >> 5, n = idx & 31;
    lw[k * LWS + n] = wgt[n * 256 + k];
  }
  __syncthreads();

  int wave  = tid >> 5;
  int lane  = tid & 31;
  int ml    = lane & 15;        // M (pixel) offset inside tile
  int khalf = lane >> 4;        // 0: holds K={0,1}, 1: holds K={2,3}
  long m0 = ((long)blockIdx.x * 4 + wave) * 16;   // 16-pixel tile base
  int  b  = (int)(m0 >> 14);                      // / (H*W)
  int  pix = (int)(m0 & (HW_ - 1));
  const float* xb = x + (long)b * C_ * HW_ + pix + ml;

  v8f acc0 = {};   // N-tile 0 (outputs 0..15)
  v8f acc1 = {};   // N-tile 1 (outputs 16..31)
  #pragma unroll 4
  for (int k0 = 0; k0 < 256; k0 += 4) {
    int ka = k0 + khalf * 2;
    v2f a, b0, b1;
    a.x  = xb[(long)ka * HW_];
    a.y  = xb[(long)(ka + 1) * HW_];
    b0.x = lw[ka * LWS + ml];
    b0.y = lw[(ka + 1) * LWS + ml];
    b1.x = lw[ka * LWS + 16 + ml];
    b1.y = lw[(ka + 1) * LWS + 16 + ml];
    acc0 = __builtin_amdgcn_wmma_f32_16x16x4_f32(false, a, false, b0,
                                                 (short)0, acc0, false, false);
    acc1 = __builtin_amdgcn_wmma_f32_16x16x4_f32(false, a, false, b1,
                                                 (short)0, acc1, false, false);
  }

  float bs0 = bias[ml];
  float bs1 = bias[16 + ml];
  int Mb = khalf * 8;
  #pragma unroll
  for (int r = 0; r < 8; ++r) {
    long m = m0 + Mb + r;
    off[m * NOUT_ + ml]      = acc0[r] + bs0;
    off[m * NOUT_ + 16 + ml] = acc1[r] + bs1;
  }
}

// ---------------------------------------------------------------------------
// Kernel 2: bilinear border-clamped sampling, 64 channels per group.
//   ix = w + 0.25*off_x[k] + (j? +0.25 : -0.25), k = g*4 + i*2 + j
//   block = one (b, g, oh) row; lane = ow; coalesced stores per channel.
// ---------------------------------------------------------------------------
__global__ void __launch_bounds__(256) bilinear_upsample(
    const float* __restrict__ x, const float* __restrict__ off,
    float* __restrict__ out) {
  int ow = threadIdx.x;
  int oh = blockIdx.x;
  int g  = blockIdx.y;
  int b  = blockIdx.z;
  int h = oh >> 1, ii = oh & 1;
  int w = ow >> 1, jj = ow & 1;
  int k = g * 4 + ii * 2 + jj;

  long pidx = ((long)b * HW_ + h * W_ + w) * NOUT_;
  float ix = (float)w + 0.25f * off[pidx + k]      + (jj ? 0.25f : -0.25f);
  float iy = (float)h + 0.25f * off[pidx + 16 + k] + (ii ? 0.25f : -0.25f);

  float x0f = floorf(ix), y0f = floorf(iy);
  float wx = ix - x0f, wy = iy - y0f;
  int xi0 = (int)x0f, yi0 = (int)y0f;
  int x0 = min(max(xi0,     0), W_ - 1);
  int x1 = min(max(xi0 + 1, 0), W_ - 1);
  int y0 = min(max(yi0,     0), H_ - 1);
  int y1 = min(max(yi0 + 1, 0), H_ - 1);

  float w00 = (1.f - wx) * (1.f - wy);
  float w01 = wx * (1.f - wy);
  float w10 = (1.f - wx) * wy;
  float w11 = wx * wy;
  int a00 = y0 * W_ + x0, a01 = y0 * W_ + x1;
  int a10 = y1 * W_ + x0, a11 = y1 * W_ + x1;

  const float* xp = x + ((long)b * C_ + g * 64) * HW_;
  long obase = (((long)(b * C_ + g * 64)) * (2 * H_) + oh) * (2 * W_) + ow;

  #pragma unroll 4
  for (int cc = 0; cc < 64; ++cc) {
    const float* p = xp + (long)cc * HW_;
    float v = w00 * p[a00] + w01 * p[a01] + w10 * p[a10] + w11 * p[a11];
    out[obase + (long)cc * (4 * HW_)] = v;
  }
}

extern "C" void kernel_launch(void* const* d_in, const int* in_sizes, int n_in,
                              void* d_out, int out_size, void* d_ws, size_t ws_size,
                              hipStream_t stream) {
  const float* x    = (const float*)d_in[0];   // (4,256,128,128)
  const float* wgt  = (const float*)d_in[1];   // (32,256)
  const float* bias = (const float*)d_in[2];   // (32,)
  float* out = (float*)d_out;                  // (4,256,256,256)
  float* off = (float*)d_ws;                   // needs B*H*W*32*4 = 8 MB

  // Phase 1: offset GEMM (WMMA).  65536 pixels / (4 waves * 16 px) = 1024 blocks.
  offset_gemm_wmma<<<dim3(B_ * HW_ / 64), dim3(128), 0, stream>>>(x, wgt, bias, off);

  // Phase 2: bilinear upsample.  grid = (oh=256, g=4, b=4), 256 lanes = ow.
  bilinear_upsample<<<dim3(2 * H_, 4, B_), dim3(256), 0, stream>>>(x, off, out);
}